// ConnectedComponentsPostProcessor_15006615732921
// MI455X (gfx1250) — compile-verified
//
#include <hip/hip_runtime.h>
#include <hip/hip_bf16.h>
#include <stdint.h>

// Problem constants (match reference)
#define B_    8
#define C_    4
#define H_    768
#define W_    768
#define HW_   (H_ * W_)            // 589824
#define NIMG  (B_ * (C_ - 1))      // 24 independent binary masks
#define INF_LAB 0xFFFFFFFFu
#define TILE  32                   // 32x32 LDS tiles, 768/32 = 24 exact
#define NT_   (H_ / TILE)          // 24 tiles per dim
#define LOCAL_ITERS 16             // Jacobi min-propagation steps in LDS
#define NITER 6                    // global union-find passes after local resolve

#define AS1 __attribute__((address_space(1)))
#define AS3 __attribute__((address_space(3)))

typedef int v4i __attribute__((vector_size(4 * sizeof(int))));

#if __has_builtin(__builtin_amdgcn_global_load_async_to_lds_b128)
#define ASYNC_B128 1
#else
#define ASYNC_B128 0
#endif
#if __has_builtin(__builtin_amdgcn_global_load_async_to_lds_b32)
#define ASYNC_B32 1
#else
#define ASYNC_B32 0
#endif

__device__ __forceinline__ void wait_async0() {
#if __has_builtin(__builtin_amdgcn_s_wait_asynccnt)
  __builtin_amdgcn_s_wait_asynccnt(0);
#else
  asm volatile("s_wait_asynccnt 0" ::: "memory");
#endif
}

// ---------------------------------------------------------------------------
// 0) zero per-root counts (overlaid on d_out) and per-image best keys
// ---------------------------------------------------------------------------
__global__ void k_zero(unsigned* __restrict__ counts,
                       unsigned long long* __restrict__ best) {
  unsigned gid = blockIdx.x * 256u + threadIdx.x;     // grid covers NIMG*HW_ exactly
  if (gid < NIMG) best[gid] = 0ull;
  counts[gid] = 0u;
}

// ---------------------------------------------------------------------------
// 1) fused per-tile kernel: async-stage pred tile -> LDS, argmax over C=4,
//    then for each class 1..3: init labels (global raster index / INF) and run
//    LOCAL_ITERS of Jacobi min-propagation fully inside LDS. Writes labels
//    with the invariant L[p] <= p, roots self-pointing -> valid UF forest.
// ---------------------------------------------------------------------------
__global__ void k_local(const float* __restrict__ pred,
                        unsigned* __restrict__ labels) {
  __shared__ __align__(16) float    spred[C_][TILE * TILE];  // 16 KB
  __shared__ unsigned               sam[TILE * TILE];        //  4 KB
  __shared__ unsigned               slab[TILE * TILE];       //  4 KB

  const int t   = threadIdx.x;          // 0..255, owns pixels 4t..4t+3 (same row)
  const int ly  = t >> 3;               // 0..31
  const int lx0 = (t & 7) << 2;         // 0,4,...,28
  const int k0  = ly * TILE + lx0;

  const int tx = blockIdx.x, ty = blockIdx.y, b = blockIdx.z;
  const int gy  = ty * TILE + ly;
  const int gx0 = tx * TILE + lx0;
  const unsigned p0 = (unsigned)(gy * W_ + gx0);   // global raster index of pixel j=0

  const float* pb = pred + (size_t)b * C_ * HW_;

  // ---- stage 4 channels x 4 own pixels into LDS (async on gfx1250) --------
#if ASYNC_B128
  #pragma unroll
  for (int c = 0; c < C_; ++c) {
    __builtin_amdgcn_global_load_async_to_lds_b128(
        (AS1 v4i*)(pb + (size_t)c * HW_ + p0),
        (AS3 v4i*)&spred[c][k0], /*offset=*/0, /*cpol=*/0);
  }
  wait_async0();                        // own lane's LDS writes complete
#elif ASYNC_B32
  #pragma unroll
  for (int c = 0; c < C_; ++c)
    #pragma unroll
    for (int j = 0; j < 4; ++j)
      __builtin_amdgcn_global_load_async_to_lds_b32(
          (AS1 int*)(pb + (size_t)c * HW_ + p0 + j),
          (AS3 int*)&spred[c][k0 + j], 0, 0);
  wait_async0();
#else
  #pragma unroll
  for (int c = 0; c < C_; ++c)
    #pragma unroll
    for (int j = 0; j < 4; ++j)
      spred[c][k0 + j] = pb[(size_t)c * HW_ + p0 + j];
#endif

  // ---- argmax over channels (strict '>' => first max, matches jnp.argmax) --
  #pragma unroll
  for (int j = 0; j < 4; ++j) {
    float mv = spred[0][k0 + j];
    int am = 0;
    #pragma unroll
    for (int c = 1; c < C_; ++c) {
      float vv = spred[c][k0 + j];
      if (vv > mv) { mv = vv; am = c; }
    }
    sam[k0 + j] = (unsigned)am;
  }
  // sam written by owner thread only and read by owner thread only below.

  // ---- per-class: init + Jacobi min-propagation in LDS ---------------------
  for (int cls = 1; cls < C_; ++cls) {
    #pragma unroll
    for (int j = 0; j < 4; ++j)
      slab[k0 + j] = (sam[k0 + j] == (unsigned)cls) ? (p0 + j) : INF_LAB;
    __syncthreads();

    for (int it = 0; it < LOCAL_ITERS; ++it) {
      unsigned nv[4];
      #pragma unroll
      for (int j = 0; j < 4; ++j) {
        const int k = k0 + j;
        const int lx = lx0 + j;
        unsigned cur = slab[k];
        if (cur != INF_LAB) {
          if (lx > 0)        cur = min(cur, slab[k - 1]);
          if (lx < TILE - 1) cur = min(cur, slab[k + 1]);
          if (ly > 0)        cur = min(cur, slab[k - TILE]);
          if (ly < TILE - 1) cur = min(cur, slab[k + TILE]);
        }
        nv[j] = cur;
      }
      __syncthreads();
      #pragma unroll
      for (int j = 0; j < 4; ++j) slab[k0 + j] = nv[j];
      __syncthreads();
    }

    // write back own 4 contiguous labels (16B aligned)
    unsigned* Li = labels + (size_t)(b * (C_ - 1) + (cls - 1)) * HW_;
    uint4 out4 = make_uint4(slab[k0], slab[k0 + 1], slab[k0 + 2], slab[k0 + 3]);
    *(uint4*)&Li[p0] = out4;
    __syncthreads();                    // protect slab reuse for next class
  }
}

// ---------------------------------------------------------------------------
// 2) union-find merge: hook larger root under smaller via atomicMin (4-conn)
// ---------------------------------------------------------------------------
__global__ void k_merge(unsigned* __restrict__ labels) {
  const unsigned gid = blockIdx.x * 256u + threadIdx.x;   // < NIMG*HW_
  const unsigned img = gid / HW_;
  const unsigned p   = gid - img * HW_;
  unsigned* L = labels + (size_t)img * HW_;

  const unsigned lp = L[p];
  if (lp == INF_LAB) return;

  const unsigned x = p % W_;
  unsigned m = INF_LAB;                                   // bg neighbors are INF: no effect
  if (x > 0)          m = min(m, L[p - 1]);
  if (x < W_ - 1u)    m = min(m, L[p + 1]);
  if (p >= (unsigned)W_)        m = min(m, L[p - W_]);
  if (p < (unsigned)(HW_ - W_)) m = min(m, L[p + W_]);

  if (m < lp) {
    unsigned r1 = lp;
    { unsigned n = L[r1]; while (n != r1) { r1 = n; n = L[r1]; } }
    unsigned r2 = m;
    { unsigned n = L[r2]; while (n != r2) { r2 = n; n = L[r2]; } }
    if (r1 == r2) return;
    unsigned hi = r1 > r2 ? r1 : r2;
    unsigned lo = r1 > r2 ? r2 : r1;
    atomicMin(&L[hi], lo);
  }
}

// ---------------------------------------------------------------------------
// 3) path compression: label <- root(label)
// ---------------------------------------------------------------------------
__global__ void k_compress(unsigned* __restrict__ labels) {
  const unsigned gid = blockIdx.x * 256u + threadIdx.x;
  const unsigned img = gid / HW_;
  const unsigned p   = gid - img * HW_;
  unsigned* L = labels + (size_t)img * HW_;

  const unsigned lp = L[p];
  if (lp == INF_LAB) return;
  unsigned r = lp, n = L[r];
  while (n != r) { r = n; n = L[r]; }
  if (r != lp) L[p] = r;
}

// ---------------------------------------------------------------------------
// 4) component sizes: histogram of roots
// ---------------------------------------------------------------------------
__global__ void k_count(const unsigned* __restrict__ labels,
                        unsigned* __restrict__ counts) {
  const unsigned gid = blockIdx.x * 256u + threadIdx.x;
  const unsigned img = gid / HW_;
  const unsigned lp = labels[gid];
  if (lp == INF_LAB) return;
  atomicAdd(&counts[(size_t)img * HW_ + lp], 1u);
}

// ---------------------------------------------------------------------------
// 5) per-image best component: max size, ties -> smallest root
//    key = (count << 32) | (~root), reduced with atomicMax
// ---------------------------------------------------------------------------
__global__ void k_best(const unsigned* __restrict__ counts,
                       unsigned long long* __restrict__ best) {
  const unsigned gid = blockIdx.x * 256u + threadIdx.x;
  const unsigned img = gid / HW_;
  const unsigned x   = gid - img * HW_;
  const unsigned cnt = counts[gid];
  if (cnt == 0u) return;
  const unsigned long long key =
      ((unsigned long long)cnt << 32) | (unsigned long long)(INF_LAB - x);
  atomicMax(&best[img], key);
}

// ---------------------------------------------------------------------------
// 6) write output: fg channels = (label == best root), bg = 1 - clip(sum,0,1)
// ---------------------------------------------------------------------------
__global__ void k_final(const unsigned* __restrict__ labels,
                        const unsigned long long* __restrict__ best,
                        float* __restrict__ out) {
  const unsigned p = blockIdx.x * 256u + threadIdx.x;
  const int b = blockIdx.y;

  float s = 0.f;
  float f[C_ - 1];
  #pragma unroll
  for (int c = 0; c < C_ - 1; ++c) {
    const int i = b * (C_ - 1) + c;
    const unsigned lp = labels[(size_t)i * HW_ + p];
    const unsigned long long bk = best[i];
    const unsigned root = INF_LAB - (unsigned)(bk & 0xFFFFFFFFull);
    const float v = (bk != 0ull && lp == root) ? 1.f : 0.f;
    f[c] = v;
    s += v;
  }
  float* ob = out + (size_t)b * C_ * HW_;
  ob[p] = 1.f - fminf(s, 1.f);
  #pragma unroll
  for (int c = 0; c < C_ - 1; ++c) ob[(size_t)(c + 1) * HW_ + p] = f[c];
}

// ---------------------------------------------------------------------------
extern "C" void kernel_launch(void* const* d_in, const int* in_sizes, int n_in,
                              void* d_out, int out_size, void* d_ws, size_t ws_size,
                              hipStream_t stream) {
  (void)in_sizes; (void)n_in; (void)out_size; (void)ws_size;

  const float* pred = (const float*)d_in[0];
  float* out = (float*)d_out;

  // workspace layout: [ labels : NIMG*HW u32 ][ best : NIMG u64 ]
  unsigned* labels = (unsigned*)d_ws;
  unsigned long long* best =
      (unsigned long long*)((char*)d_ws + (size_t)NIMG * HW_ * sizeof(unsigned));
  // per-root counts overlay the front of d_out (56.6MB of 75.5MB);
  // zeroed each call, fully overwritten by k_final afterwards.
  unsigned* counts = (unsigned*)d_out;

  const dim3 blk(256);
  const dim3 gridTile(NT_, NT_, B_);            // 24 x 24 x 8 tiles
  const dim3 gridPix(HW_ / 256, B_);            // 2304 x 8, exact
  const int  gridAll = (NIMG * HW_) / 256;      // 55296, exact

  k_zero<<<gridAll, blk, 0, stream>>>(counts, best);
  k_local<<<gridTile, blk, 0, stream>>>(pred, labels);
  for (int it = 0; it < NITER; ++it) {
    k_merge<<<gridAll, blk, 0, stream>>>(labels);
    k_compress<<<gridAll, blk, 0, stream>>>(labels);
  }
  k_count<<<gridAll, blk, 0, stream>>>(labels, counts);
  k_best<<<gridAll, blk, 0, stream>>>(counts, best);
  k_final<<<gridPix, blk, 0, stream>>>(labels, best, out);
}